// GlobalContextBlock1d_70403103916045
// MI455X (gfx1250) — compile-verified
//
#include <hip/hip_runtime.h>
#include <hip/hip_bf16.h>

// GlobalContextBlock1d for MI455X (gfx1250), wave32.
// B=32, C=256, L=8192, R=16 -> CR=16.
//
// Pass 1 (k_score_ctx): fused scores+softmax-numerator+context-pooling.
//   One block per (batch, 256-length chunk). Caches the 256x256 x-tile in
//   LDS (258 KB, CDNA5 has 320 KB/WGP), computes e_l = exp(score_l + b) and
//   the per-chunk partial ctx_num[c] = sum_l x[c,l]*e_l and partial exp-sum.
//   Deterministic: partials go to private workspace slots, no atomics.
// Pass 2 (k_mlp): one block. Reduces partials, normalizes ctx, then runs the
//   bottleneck MLP with V_WMMA_F32_16X16X4_F32 (fp32 WMMA keeps reference
//   precision): GEMM1 16x32x256, in-register LayerNorm+ReLU via shfl_xor(16),
//   GEMM2 256x32x16. Result t2[b][c] written to workspace.
// Pass 3 (k_residual): out = x + t2 broadcast, float4-vectorized streaming.

typedef float v2f __attribute__((ext_vector_type(2)));
typedef float v8f __attribute__((ext_vector_type(8)));

#define B_      32
#define C_      256
#define L_      8192
#define CR_     16
#define NCHUNK  32
#define LT      256          // L_/NCHUNK
#define LN_EPS  1e-5f

// ---------------------------------------------------------------------------
// Kernel 1: fused score + exp + partial context pooling.
// grid = (NCHUNK, B_), block = 256 threads (8 waves).
// dynamic LDS: xs[C_][LT+1] tile + ev[LT] + red[8]
// ---------------------------------------------------------------------------
__global__ void __launch_bounds__(256)
k_score_ctx(const float* __restrict__ x,
            const float* __restrict__ w_score,
            const float* __restrict__ b_score,
            float* __restrict__ pctx,    // [NCHUNK][B][C]
            float* __restrict__ psum)    // [NCHUNK][B]
{
    extern __shared__ float smem[];
    float* xs  = smem;                    // [C_][LT+1] padded rows
    float* ev  = smem + C_ * (LT + 1);    // [LT]
    float* red = ev + LT;                 // [8]

    const int tid   = threadIdx.x;        // == l within chunk
    const int chunk = blockIdx.x;
    const int b     = blockIdx.y;
    const int l0    = chunk * LT;
    const float* xb = x + (size_t)b * C_ * L_;

    // Phase 1: stream the tile into LDS, accumulate this l's score.
    float score = 0.f;
#pragma unroll 8
    for (int c = 0; c < C_; ++c) {
        float val = xb[(size_t)c * L_ + l0 + tid];
        xs[c * (LT + 1) + tid] = val;
        score = fmaf(w_score[c], val, score);
    }
    float e = expf(score + b_score[0]);   // scores ~N(0,0.1): exp is safe
    ev[tid] = e;

    // Block-reduce sum(e) -> psum[chunk][b]
    float s = e;
    for (int off = 16; off > 0; off >>= 1) s += __shfl_down(s, off, 32);
    if ((tid & 31) == 0) red[tid >> 5] = s;
    __syncthreads();
    if (tid == 0) {
        float tot = 0.f;
        for (int w = 0; w < 8; ++w) tot += red[w];
        psum[chunk * B_ + b] = tot;
    }

    // Phase 2: thread tid owns channel c = tid; padded stride (LT+1) makes
    // xs[tid*(LT+1)+j] conflict-free across the wave.
    float acc = 0.f;
#pragma unroll 8
    for (int j = 0; j < LT; ++j)
        acc = fmaf(xs[tid * (LT + 1) + j], ev[j], acc);
    pctx[((size_t)chunk * B_ + b) * C_ + tid] = acc;
}

// ---------------------------------------------------------------------------
// Kernel 2: reduce partials, normalize, bottleneck MLP via fp32 WMMA.
// grid = 1 block of 512 threads (16 waves).
// ---------------------------------------------------------------------------
__global__ void __launch_bounds__(512)
k_mlp(const float* __restrict__ pctx, const float* __restrict__ psum,
      const float* __restrict__ w1,  const float* __restrict__ b1,
      const float* __restrict__ lnw, const float* __restrict__ lnb,
      const float* __restrict__ w2,  const float* __restrict__ b2,
      float* __restrict__ t2)        // [B][C]
{
    __shared__ float ctx[B_ * (C_ + 1)];   // [b][c], stride 257 (bank-safe)
    __shared__ float zz[B_];
    __shared__ float tl[CR_ * 33];         // [r][b], stride 33 (bank-safe)

    const int tid = threadIdx.x;

    // Z_b = sum over chunks of psum
    if (tid < B_) {
        float s = 0.f;
        for (int k = 0; k < NCHUNK; ++k) s += psum[k * B_ + tid];
        zz[tid] = s;
    }
    __syncthreads();

    // ctx[b][c] = (sum_k pctx[k][b][c]) / Z_b
    for (int i = tid; i < B_ * C_; i += 512) {
        int b = i >> 8, c = i & (C_ - 1);
        float s = 0.f;
        for (int k = 0; k < NCHUNK; ++k)
            s += pctx[((size_t)k * B_ + b) * C_ + c];
        ctx[b * (C_ + 1) + c] = s / zz[b];
    }
    __syncthreads();

    const int wave = tid >> 5;
    const int lane = tid & 31;
    const int n    = lane & 15;     // WMMA column (and A-row) index
    const int hi   = lane >> 4;     // K-pair / M-half selector

    // ---- GEMM1: t[16r x 32b] = w1[16,256] * ctx^T, waves 0,1 (n-tiles) ----
    if (wave < 2) {
        const int n0 = wave * 16;
        v8f acc = {};
        for (int k0 = 0; k0 < C_; k0 += 4) {
            v2f a, bm;
            // A 16x4 fp32 layout: lane%16 = M, lane/16 picks K-pair {0,1}|{2,3}
            a.x = w1[n * C_ + k0 + 2 * hi];
            a.y = w1[n * C_ + k0 + 2 * hi + 1];
            // B 4x16: B[k][col] = ctx[b = n0+col][c = k0+k]
            bm.x = ctx[(n0 + n) * (C_ + 1) + k0 + 2 * hi];
            bm.y = ctx[(n0 + n) * (C_ + 1) + k0 + 2 * hi + 1];
            acc = __builtin_amdgcn_wmma_f32_16x16x4_f32(
                      false, a, false, bm, (short)0, acc, false, false);
        }
        // D layout: acc[v] = t_pre[r = v+8*hi][b = n0+n]. Bias + LayerNorm
        // over r (16 values split across lane pair L and L^16) + ReLU.
        float tv[8];
        float lsum = 0.f, lsq = 0.f;
#pragma unroll
        for (int v = 0; v < 8; ++v) {
            int r = v + 8 * hi;
            float t = acc[v] + b1[r];
            tv[v] = t;
            lsum += t;
            lsq  += t * t;
        }
        lsum += __shfl_xor(lsum, 16, 32);
        lsq  += __shfl_xor(lsq, 16, 32);
        float mu  = lsum * (1.f / 16.f);
        float var = lsq * (1.f / 16.f) - mu * mu;
        float rs  = rsqrtf(var + LN_EPS);
#pragma unroll
        for (int v = 0; v < 8; ++v) {
            int r = v + 8 * hi;
            float t = (tv[v] - mu) * rs * lnw[r] + lnb[r];
            tl[r * 33 + n0 + n] = fmaxf(t, 0.f);
        }
    }
    __syncthreads();

    // ---- GEMM2: t2[256c x 32b] = w2[256,16] * t[16,32]; 16 waves x 2 tiles --
    const int m0 = wave * 16;
    for (int nt = 0; nt < 2; ++nt) {
        const int n0 = nt * 16;
        v8f acc = {};
#pragma unroll
        for (int k0 = 0; k0 < CR_; k0 += 4) {
            v2f a, bm;
            a.x = w2[(m0 + n) * CR_ + k0 + 2 * hi];
            a.y = w2[(m0 + n) * CR_ + k0 + 2 * hi + 1];
            bm.x = tl[(k0 + 2 * hi) * 33 + n0 + n];
            bm.y = tl[(k0 + 2 * hi + 1) * 33 + n0 + n];
            acc = __builtin_amdgcn_wmma_f32_16x16x4_f32(
                      false, a, false, bm, (short)0, acc, false, false);
        }
#pragma unroll
        for (int v = 0; v < 8; ++v) {
            int c = m0 + v + 8 * hi;
            t2[(n0 + n) * C_ + c] = acc[v] + b2[c];
        }
    }
}

// ---------------------------------------------------------------------------
// Kernel 3: out[b,c,:] = x[b,c,:] + t2[b,c].  grid = B*C rows, float4 stream.
// ---------------------------------------------------------------------------
__global__ void __launch_bounds__(256)
k_residual(const float* __restrict__ x, const float* __restrict__ t2,
           float* __restrict__ out)
{
    const int bc = blockIdx.x;               // b*C + c, matches t2 layout
    const float add = t2[bc];
    const size_t base = (size_t)bc * L_;
    const float4* xi = (const float4*)(x + base);
    float4* xo = (float4*)(out + base);
    for (int i = threadIdx.x; i < L_ / 4; i += 256) {
        float4 v = xi[i];
        v.x += add; v.y += add; v.z += add; v.w += add;
        xo[i] = v;
    }
}

// ---------------------------------------------------------------------------
extern "C" void kernel_launch(void* const* d_in, const int* in_sizes, int n_in,
                              void* d_out, int out_size, void* d_ws, size_t ws_size,
                              hipStream_t stream)
{
    const float* x       = (const float*)d_in[0];
    const float* w_score = (const float*)d_in[1];
    const float* b_score = (const float*)d_in[2];
    const float* w1      = (const float*)d_in[3];
    const float* b1      = (const float*)d_in[4];
    const float* lnw     = (const float*)d_in[5];
    const float* lnb     = (const float*)d_in[6];
    const float* w2      = (const float*)d_in[7];
    const float* b2      = (const float*)d_in[8];
    float* out = (float*)d_out;

    float* ws   = (float*)d_ws;
    float* pctx = ws;                          // NCHUNK*B*C   = 262144 floats
    float* psum = pctx + NCHUNK * B_ * C_;     // NCHUNK*B     = 1024 floats
    float* t2   = psum + NCHUNK * B_;          // B*C          = 8192 floats
    // total ~1.06 MB of workspace; every element fully written each call.

    const size_t smem1 = (size_t)(C_ * (LT + 1) + LT + 8) * sizeof(float); // ~258 KB

    k_score_ctx<<<dim3(NCHUNK, B_), 256, smem1, stream>>>(x, w_score, b_score,
                                                          pctx, psum);
    k_mlp<<<1, 512, 0, stream>>>(pctx, psum, w1, b1, lnw, lnb, w2, b2, t2);
    k_residual<<<B_ * C_, 256, 0, stream>>>(x, t2, out);
}